// CircuitChannel_42795054137472
// MI455X (gfx1250) — compile-verified
//
#include <hip/hip_runtime.h>
#include <math.h>

#define NQ   20
#define DIMS (1 << NQ)      // 2^20 amplitudes per batch element
#define BTCH 8

typedef float v8f __attribute__((ext_vector_type(8)));
typedef float v2f __attribute__((ext_vector_type(2)));

// ---------------------------------------------------------------------------
// Interleave psi_re / psi_im into the working state (d_out) as float2.
// ---------------------------------------------------------------------------
__global__ void k_init(const float* __restrict__ re, const float* __restrict__ im,
                       float* __restrict__ st, int n) {
  int i = blockIdx.x * blockDim.x + threadIdx.x;
  if (i < n) {
    v2f o; o.x = re[i]; o.y = im[i];
    *(v2f*)(st + 2 * (size_t)i) = o;
  }
}

// ---------------------------------------------------------------------------
// Build fused 4-qubit group unitaries: for layer L, group g (qubits 4g..4g+3),
// U[m][k] = prod_s RX(theta[L][4g+s])[bit_{3-s}(m)][bit_{3-s}(k)]
// RX entries: diag = cos(t/2) (real), offdiag = -i*sin(t/2).
// Layout in ws: mats[(L*5+g)*512 + {0:re,256:im} + m*16 + k]
// ---------------------------------------------------------------------------
__global__ void k_mats(const float* __restrict__ thetas, float* __restrict__ mats) {
  int e = blockIdx.x * blockDim.x + threadIdx.x;
  if (e >= 4 * 5 * 256) return;
  int layer = e / 1280;
  int rem   = e % 1280;
  int g     = rem / 256;
  int mi    = rem % 256;
  int m = mi >> 4, k = mi & 15;
  float ar = 1.f, ai = 0.f;
  for (int s = 0; s < 4; ++s) {
    float th = thetas[layer * NQ + 4 * g + s];
    float c  = cosf(0.5f * th);
    float sn = sinf(0.5f * th);
    int bm = (m >> (3 - s)) & 1;
    int bk = (k >> (3 - s)) & 1;
    float er, ei;
    if (bm == bk) { er = c;   ei = 0.f; }
    else          { er = 0.f; ei = -sn; }
    float nr = ar * er - ai * ei;
    float ni = ar * ei + ai * er;
    ar = nr; ai = ni;
  }
  float* base = mats + (size_t)(layer * 5 + g) * 512;
  base[m * 16 + k]       = ar;
  base[256 + m * 16 + k] = ai;
}

// ---------------------------------------------------------------------------
// Apply one 16x16 complex group unitary along the 4-bit field at bit P of the
// state index.  state idx = (hi << (P+4)) | (j << P) | lo, j in 0..15.
// Each wave computes 16x16-output tiles with V_WMMA_F32_16X16X4_F32:
//   Dr = Ur*Br + (-Ui)*Bi ;  Di = Ur*Bi + Ui*Br   (16 WMMAs per tile)
// VGPR layouts per ISA 7.12.2:
//   A[m][k]:  lane = (m) + 16*(k>>1), reg = k&1      (m = lane&15)
//   B[k][n]:  lane = (n) + 16*(k>>1 within set), reg = k&1
//   D[m][n]:  reg = m&7, lane = n + 16*(m>>3)
// SIGN: fuse the CZ-ring diagonal sign (end of layer).
// ---------------------------------------------------------------------------
template <int P, bool SIGN>
__global__ __launch_bounds__(256) void k_pass(float* __restrict__ st,
                                              const float* __restrict__ matR,
                                              const float* __restrict__ matI) {
  const int lane   = threadIdx.x & 31;
  const int laneHi = lane >> 4;     // 0 or 1
  const int laneLo = lane & 15;
  const int wave   = blockIdx.x * (blockDim.x >> 5) + (threadIdx.x >> 5);
  const int nwav   = gridDim.x * (blockDim.x >> 5);

  // Load A operands (Ur, Ui) once; build -Ui for the subtraction term.
  v2f aR[4], aI[4], aM[4];
#pragma unroll
  for (int t = 0; t < 4; ++t) {
    const int k0 = 4 * t + 2 * laneHi;
    aR[t].x =  matR[laneLo * 16 + k0];
    aR[t].y =  matR[laneLo * 16 + k0 + 1];
    aI[t].x =  matI[laneLo * 16 + k0];
    aI[t].y =  matI[laneLo * 16 + k0 + 1];
    aM[t].x = -aI[t].x;
    aM[t].y = -aI[t].y;
  }

  const int TILES = BTCH * (DIMS / 256);  // 16 rows x 16 cols per tile = 32768
  for (int T = wave; T < TILES; T += nwav) {
    const int b    = T >> 12;                          // batch element
    const int c    = ((T & 4095) << 4) | laneLo;       // this lane's column
    const int lo   = c & ((1 << P) - 1);
    const int hi   = c >> P;
    const int base = (hi << (P + 4)) | lo;             // idx at j = 0
    float* bp = st + (((size_t)b << NQ) + (size_t)base) * 2;

    // Gather B: complex state[j][n], j = 4t + v + 2*laneHi, n = laneLo.
    float* lp = bp + (((size_t)(2 * laneHi)) << P) * 2;
    v2f b0[4], b1[4];
#pragma unroll
    for (int t = 0; t < 4; ++t) {
      b0[t] = *(const v2f*)(lp + (((size_t)(4 * t + 0)) << P) * 2);
      b1[t] = *(const v2f*)(lp + (((size_t)(4 * t + 1)) << P) * 2);
    }
    v2f bR[4], bI[4];
#pragma unroll
    for (int t = 0; t < 4; ++t) {
      bR[t].x = b0[t].x; bR[t].y = b1[t].x;
      bI[t].x = b0[t].y; bI[t].y = b1[t].y;
    }

    v8f dR = {}; v8f dI = {};
#pragma unroll
    for (int t = 0; t < 4; ++t) {
      dR = __builtin_amdgcn_wmma_f32_16x16x4_f32(false, aR[t], false, bR[t],
                                                 (short)0, dR, false, false);
      dI = __builtin_amdgcn_wmma_f32_16x16x4_f32(false, aR[t], false, bI[t],
                                                 (short)0, dI, false, false);
    }
#pragma unroll
    for (int t = 0; t < 4; ++t) {
      dR = __builtin_amdgcn_wmma_f32_16x16x4_f32(false, aM[t], false, bI[t],
                                                 (short)0, dR, false, false);
      dI = __builtin_amdgcn_wmma_f32_16x16x4_f32(false, aI[t], false, bR[t],
                                                 (short)0, dI, false, false);
    }

    // Scatter D: element (m = r + 8*laneHi, n = laneLo).
    float* sp = bp + (((size_t)(8 * laneHi)) << P) * 2;
#pragma unroll
    for (int r = 0; r < 8; ++r) {
      float re = dR[r], im = dI[r];
      if (SIGN) {
        const int m   = r + 8 * laneHi;
        const int idx = base + (m << P);
        if (__popc(idx & (idx >> 1)) & 1) { re = -re; im = -im; }
      }
      v2f o; o.x = re; o.y = im;
      *(v2f*)(sp + (((size_t)r) << P) * 2) = o;
    }
  }
}

// ---------------------------------------------------------------------------
// Measurement on qubit 0 (bit 19): p0[b] = sum_{idx < 2^19} |psi[b][idx]|^2.
// Deterministic two-stage reduction, then elementwise zero/rescale.
// ---------------------------------------------------------------------------
__global__ __launch_bounds__(256) void k_meas_partial(const float* __restrict__ st,
                                                      float* __restrict__ partial) {
  __shared__ float red[256];
  const int b = blockIdx.x >> 6;        // 8 batches x 64 chunks
  const int w = blockIdx.x & 63;
  const int per = (DIMS / 2) / 64;      // 8192 complex elems per block
  const float* p = st + ((size_t)b << (NQ + 1));
  float s = 0.f;
  for (int i = threadIdx.x; i < per; i += 256) {
    int idx = w * per + i;
    float re = p[2 * idx], im = p[2 * idx + 1];
    s += re * re + im * im;
  }
  red[threadIdx.x] = s;
  __syncthreads();
  for (int k = 128; k > 0; k >>= 1) {
    if (threadIdx.x < k) red[threadIdx.x] += red[threadIdx.x + k];
    __syncthreads();
  }
  if (threadIdx.x == 0) partial[b * 64 + w] = red[0];
}

__global__ void k_meas_final(const float* __restrict__ partial,
                             const float* __restrict__ u,
                             float* __restrict__ factors) {
  __shared__ float red[64];
  const int b = blockIdx.x;
  red[threadIdx.x] = partial[b * 64 + threadIdx.x];
  __syncthreads();
  for (int k = 32; k > 0; k >>= 1) {
    if (threadIdx.x < k) red[threadIdx.x] += red[threadIdx.x + k];
    __syncthreads();
  }
  if (threadIdx.x == 0) {
    float p0 = red[0];
    int   m  = (u[b] >= p0) ? 1 : 0;
    float p  = m ? (1.f - p0) : p0;
    float f  = 1.f / sqrtf(p);
    factors[2 * b + 0] = (m == 0) ? f : 0.f;
    factors[2 * b + 1] = (m == 1) ? f : 0.f;
  }
}

__global__ void k_meas_apply(float* __restrict__ st, const float* __restrict__ factors) {
  int i = blockIdx.x * blockDim.x + threadIdx.x;   // over 8M complex elems
  int b    = i >> NQ;
  int half = (i >> (NQ - 1)) & 1;
  float f  = factors[2 * b + half];
  v2f v = *(v2f*)(st + 2 * (size_t)i);
  v.x *= f; v.y *= f;
  *(v2f*)(st + 2 * (size_t)i) = v;
}

// ---------------------------------------------------------------------------
extern "C" void kernel_launch(void* const* d_in, const int* in_sizes, int n_in,
                              void* d_out, int out_size, void* d_ws, size_t ws_size,
                              hipStream_t stream) {
  (void)in_sizes; (void)n_in; (void)out_size; (void)ws_size;
  const float* psi_re = (const float*)d_in[0];
  const float* psi_im = (const float*)d_in[1];
  const float* thetas = (const float*)d_in[2];   // [4, 20]
  const float* u      = (const float*)d_in[3];   // [8]

  float* st      = (float*)d_out;                // working state, interleaved
  float* ws      = (float*)d_ws;
  float* mats    = ws;                           // 4*5*512 = 10240 floats... (20 mats x 512)
  float* partial = ws + 4 * 5 * 512;             // 8*64 floats
  float* factors = partial + 8 * 64;             // 16 floats

  const int NTOT = BTCH * DIMS;
  k_init<<<(NTOT + 255) / 256, 256, 0, stream>>>(psi_re, psi_im, st, NTOT);
  k_mats<<<(4 * 5 * 256 + 127) / 128, 128, 0, stream>>>(thetas, mats);

  const int PBLK = 512, PTHR = 256;
  for (int layer = 0; layer < 4; ++layer) {
    float* mb = mats + (size_t)layer * 5 * 512;
    k_pass<16, false><<<PBLK, PTHR, 0, stream>>>(st, mb + 0 * 512, mb + 0 * 512 + 256);
    k_pass<12, false><<<PBLK, PTHR, 0, stream>>>(st, mb + 1 * 512, mb + 1 * 512 + 256);
    k_pass< 8, false><<<PBLK, PTHR, 0, stream>>>(st, mb + 2 * 512, mb + 2 * 512 + 256);
    k_pass< 4, false><<<PBLK, PTHR, 0, stream>>>(st, mb + 3 * 512, mb + 3 * 512 + 256);
    k_pass< 0, true ><<<PBLK, PTHR, 0, stream>>>(st, mb + 4 * 512, mb + 4 * 512 + 256);
  }

  k_meas_partial<<<8 * 64, 256, 0, stream>>>(st, partial);
  k_meas_final<<<8, 64, 0, stream>>>(partial, u, factors);
  k_meas_apply<<<NTOT / 256, 256, 0, stream>>>(st, factors);
}